// VQSimpleMLPSlotLatentAction_6390911337113
// MI455X (gfx1250) — compile-verified
//
#include <hip/hip_runtime.h>
#include <hip/hip_bf16.h>

typedef float v2f __attribute__((ext_vector_type(2)));
typedef float v8f __attribute__((ext_vector_type(8)));

constexpr int BB = 128, NN = 32, SS = 12;
constexpr int SLOT = 256, EMB = 256, HID = 1024, ACT = 128, NCODE = 1024;
constexpr int TTOK = BB * NN * SS;          // 49152 encoder tokens
constexpr int MROW = BB * (NN - 1) * SS;    // 47616 action rows

// d_out layout (flat, reference tuple order, all float32)
constexpr size_t SZ_AD  = (size_t)BB * (NN - 1) * 2 * SS * ACT;   // action_dist
constexpr size_t SZ_P   = (size_t)MROW * ACT;                     // per-plane
constexpr size_t OFF_AD  = 0;
constexpr size_t OFF_Z   = OFF_AD + SZ_AD;
constexpr size_t OFF_AV  = OFF_Z + SZ_P;
constexpr size_t OFF_QST = OFF_AV + SZ_P;
constexpr size_t OFF_IDX = OFF_QST + SZ_P;
constexpr size_t OFF_QL  = OFF_IDX + (size_t)MROW;
constexpr size_t OFF_CL  = OFF_QL + 1;

__device__ __forceinline__ v8f wmma_f32_k4(v2f a, v2f b, v8f c) {
  // D = A(16x4,f32) * B(4x16,f32) + C(16x16,f32)  -> v_wmma_f32_16x16x4_f32
  return __builtin_amdgcn_wmma_f32_16x16x4_f32(false, a, false, b, (short)0, c,
                                               false, false);
}

__global__ void init_loss_kernel(float* out) {
  if (threadIdx.x == 0) { out[OFF_QL] = 0.0f; out[OFF_CL] = 0.0f; }
}

// ---------------------------------------------------------------------------
// Encoder: per 16-token tile: Linear(256->1024)+ReLU, LayerNorm,
// Linear(1024->256), then mean/var heads (256->128) into workspace.
// 256 threads = 8 wave32.
// ---------------------------------------------------------------------------
__global__ __launch_bounds__(256) void enc_kernel(
    const float* __restrict__ slots,
    const float* __restrict__ W1, const float* __restrict__ b1,
    const float* __restrict__ ln_g, const float* __restrict__ ln_b,
    const float* __restrict__ W2, const float* __restrict__ b2,
    const float* __restrict__ Wm, const float* __restrict__ bm,
    const float* __restrict__ Wv, const float* __restrict__ bv,
    float* __restrict__ mean_tok, float* __restrict__ var_tok) {
  __shared__ float lds_ae[16 * SLOT];  // slots tile, later reused for e tile
  __shared__ float lds_h[16 * HID];    // hidden activations (64 KB)

  const int tid  = threadIdx.x;
  const int wave = tid >> 5;
  const int lane = tid & 31;
  const int sub  = lane & 15;  // N / M index within fragment
  const int half = lane >> 4;  // 0 or 1
  const int kb   = half * 2;   // K offset carried by this half-wave
  const int t0   = blockIdx.x * 16;

  // --- load 16x256 slots tile (contiguous) ---
  {
    const float4* src = (const float4*)(slots + (size_t)t0 * SLOT);
    float4* dst = (float4*)lds_ae;
    for (int i = tid; i < 16 * SLOT / 4; i += 256) dst[i] = src[i];
  }
  __syncthreads();

  // --- GEMM1: h = relu(slots @ W1 + b1) ; each wave -> 8 column tiles ---
  for (int nt = wave * 8; nt < wave * 8 + 8; ++nt) {
    const int n0 = nt * 16;
    const int n  = n0 + sub;
    v8f c = {0.f, 0.f, 0.f, 0.f, 0.f, 0.f, 0.f, 0.f};
#pragma unroll 8
    for (int k0 = 0; k0 < SLOT; k0 += 4) {
      v2f a, bf;
      a.x  = lds_ae[sub * SLOT + k0 + kb];
      a.y  = lds_ae[sub * SLOT + k0 + kb + 1];
      bf.x = W1[(size_t)(k0 + kb) * HID + n];
      bf.y = W1[(size_t)(k0 + kb + 1) * HID + n];
      c = wmma_f32_k4(a, bf, c);
    }
    const float bias = b1[n];
#pragma unroll
    for (int r = 0; r < 8; ++r) {
      float v = c[r] + bias;
      v = v > 0.f ? v : 0.f;
      lds_h[(r + half * 8) * HID + n] = v;
    }
  }
  __syncthreads();

  // --- LayerNorm: each 16-lane half-wave owns one row (16 rows total) ---
  {
    const int row = wave * 2 + half;
    float sum = 0.f, sq = 0.f;
    for (int j = sub; j < HID; j += 16) {
      float x = lds_h[row * HID + j];
      sum += x; sq += x * x;
    }
#pragma unroll
    for (int msk = 1; msk < 16; msk <<= 1) {
      sum += __shfl_xor(sum, msk, 16);
      sq  += __shfl_xor(sq, msk, 16);
    }
    const float mu   = sum * (1.0f / HID);
    const float var  = sq * (1.0f / HID) - mu * mu;
    const float rstd = rsqrtf(var + 1e-5f);
    for (int j = sub; j < HID; j += 16) {
      float x = lds_h[row * HID + j];
      lds_h[row * HID + j] = (x - mu) * rstd * ln_g[j] + ln_b[j];
    }
  }
  __syncthreads();

  // --- GEMM2: e = h @ W2 + b2 (reuse lds_ae as e tile) ---
  for (int nt = wave * 2; nt < wave * 2 + 2; ++nt) {
    const int n0 = nt * 16;
    const int n  = n0 + sub;
    v8f c = {0.f, 0.f, 0.f, 0.f, 0.f, 0.f, 0.f, 0.f};
#pragma unroll 8
    for (int k0 = 0; k0 < HID; k0 += 4) {
      v2f a, bf;
      a.x  = lds_h[sub * HID + k0 + kb];
      a.y  = lds_h[sub * HID + k0 + kb + 1];
      bf.x = W2[(size_t)(k0 + kb) * EMB + n];
      bf.y = W2[(size_t)(k0 + kb + 1) * EMB + n];
      c = wmma_f32_k4(a, bf, c);
    }
    const float bias = b2[n];
#pragma unroll
    for (int r = 0; r < 8; ++r)
      lds_ae[(r + half * 8) * EMB + n] = c[r] + bias;
  }
  __syncthreads();

  // --- heads: mean = e@Wm + bm ; var = |e@Wv + bv| ; wave w -> col tile w ---
  {
    const int n = wave * 16 + sub;
    v8f cm = {0.f, 0.f, 0.f, 0.f, 0.f, 0.f, 0.f, 0.f};
    v8f cv = {0.f, 0.f, 0.f, 0.f, 0.f, 0.f, 0.f, 0.f};
#pragma unroll 4
    for (int k0 = 0; k0 < EMB; k0 += 4) {
      v2f a, bmf, bvf;
      a.x   = lds_ae[sub * EMB + k0 + kb];
      a.y   = lds_ae[sub * EMB + k0 + kb + 1];
      bmf.x = Wm[(size_t)(k0 + kb) * ACT + n];
      bmf.y = Wm[(size_t)(k0 + kb + 1) * ACT + n];
      bvf.x = Wv[(size_t)(k0 + kb) * ACT + n];
      bvf.y = Wv[(size_t)(k0 + kb + 1) * ACT + n];
      cm = wmma_f32_k4(a, bmf, cm);
      cv = wmma_f32_k4(a, bvf, cv);
    }
    const float bmb = bm[n], bvb = bv[n];
#pragma unroll
    for (int r = 0; r < 8; ++r) {
      const int row = r + half * 8;
      mean_tok[(size_t)(t0 + row) * ACT + n] = cm[r] + bmb;
      var_tok[(size_t)(t0 + row) * ACT + n]  = fabsf(cv[r] + bvb);
    }
  }
}

// ---------------------------------------------------------------------------
// VQ: per 16-row tile of (B,N-1,S): build z, emit action_dist/z, WMMA score
// against the 1024-entry codebook, argmin, gather q, emit av/q_st/idx/losses.
// ---------------------------------------------------------------------------
__global__ __launch_bounds__(256) void vq_kernel(
    const float* __restrict__ noise, const float* __restrict__ codebook,
    const float* __restrict__ mean_tok, const float* __restrict__ var_tok,
    float* __restrict__ out) {
  __shared__ float lds_z[16 * ACT];   // 8 KB
  __shared__ float lds_cn[NCODE];     // 4 KB code norms
  __shared__ float red_s[16][8];
  __shared__ int   red_i[16][8];
  __shared__ int   lds_idx[16];
  __shared__ float loss_part[8];

  const int tid  = threadIdx.x;
  const int wave = tid >> 5;
  const int lane = tid & 31;
  const int sub  = lane & 15;
  const int half = lane >> 4;
  const int kb   = half * 2;
  const int m0   = blockIdx.x * 16;

  // code norms ||c||^2 (codebook is L2-resident: 512 KB)
  for (int ci = tid; ci < NCODE; ci += 256) {
    const float4* cp = (const float4*)(codebook + (size_t)ci * ACT);
    float acc = 0.f;
#pragma unroll 8
    for (int j = 0; j < ACT / 4; ++j) {
      float4 v = cp[j];
      acc += v.x * v.x + v.y * v.y + v.z * v.z + v.w * v.w;
    }
    lds_cn[ci] = acc;
  }

  // z = noise*sqrt(adv+eps)+adm ; write action_dist planes + z
  {
    const int row = tid >> 4;
    const int a0  = (tid & 15) * 8;
    const int m   = m0 + row;
    const int b   = m / ((NN - 1) * SS);
    const int rem = m % ((NN - 1) * SS);
    const int n   = rem / SS;
    const int s   = rem % SS;
    const size_t tlo = (size_t)(b * NN + n) * SS + s;
    const size_t thi = tlo + SS;
    const size_t adb = ((size_t)((b * (NN - 1) + n) * 2) * SS + s) * ACT;
#pragma unroll
    for (int j = 0; j < 8; ++j) {
      const int a = a0 + j;
      const float ml = mean_tok[tlo * ACT + a];
      const float mh = mean_tok[thi * ACT + a];
      const float vl = var_tok[tlo * ACT + a];
      const float vh = var_tok[thi * ACT + a];
      const float adm = mh - ml;
      const float adv = vh + vl;
      const float zz  = noise[(size_t)m * ACT + a] * sqrtf(adv + 1e-6f) + adm;
      lds_z[row * ACT + a]                      = zz;
      out[OFF_AD + adb + a]                     = adm;
      out[OFF_AD + adb + (size_t)SS * ACT + a]  = adv;
      out[OFF_Z + (size_t)m * ACT + a]          = zz;
    }
  }
  __syncthreads();

  // scores: ||c||^2 - 2 * (z . c) ; each wave covers 128 codes (8 tiles)
  float best_s[8];
  int   best_i[8];
#pragma unroll
  for (int r = 0; r < 8; ++r) { best_s[r] = 3.4e38f; best_i[r] = 0; }
  for (int nt = wave * 8; nt < wave * 8 + 8; ++nt) {
    const int n0 = nt * 16;
    const int n  = n0 + sub;
    v8f c = {0.f, 0.f, 0.f, 0.f, 0.f, 0.f, 0.f, 0.f};
#pragma unroll 8
    for (int k0 = 0; k0 < ACT; k0 += 4) {
      v2f a, bf;
      a.x  = lds_z[sub * ACT + k0 + kb];
      a.y  = lds_z[sub * ACT + k0 + kb + 1];
      bf.x = codebook[(size_t)n * ACT + k0 + kb];
      bf.y = codebook[(size_t)n * ACT + k0 + kb + 1];
      c = wmma_f32_k4(a, bf, c);
    }
    const float cn = lds_cn[n];
#pragma unroll
    for (int r = 0; r < 8; ++r) {
      const float sc = cn - 2.0f * c[r];
      if (sc < best_s[r]) { best_s[r] = sc; best_i[r] = n; }
    }
  }
  // reduce across the 16 lanes that share a row (rows r and r+8 per half)
#pragma unroll
  for (int r = 0; r < 8; ++r) {
    float sc = best_s[r];
    int   bi = best_i[r];
#pragma unroll
    for (int msk = 1; msk < 16; msk <<= 1) {
      const float os = __shfl_xor(sc, msk, 16);
      const int   oi = __shfl_xor(bi, msk, 16);
      if (os < sc || (os == sc && oi < bi)) { sc = os; bi = oi; }
    }
    if (sub == 0) { red_s[r + half * 8][wave] = sc; red_i[r + half * 8][wave] = bi; }
  }
  __syncthreads();
  if (tid < 16) {
    float sc = red_s[tid][0];
    int   bi = red_i[tid][0];
    for (int w = 1; w < 8; ++w) {
      const float os = red_s[tid][w];
      const int   oi = red_i[tid][w];
      if (os < sc || (os == sc && oi < bi)) { sc = os; bi = oi; }
    }
    lds_idx[tid] = bi;
    out[OFF_IDX + m0 + tid] = (float)bi;  // action_idxs
  }
  __syncthreads();

  // gather q, emit action_variability / q_st, accumulate losses
  float lsum = 0.f;
  {
    const int row = tid >> 4;
    const int a0  = (tid & 15) * 8;
    const int m   = m0 + row;
    const int qi  = lds_idx[row];
#pragma unroll
    for (int j = 0; j < 8; ++j) {
      const int a = a0 + j;
      const float q  = codebook[(size_t)qi * ACT + a];
      const float zz = lds_z[row * ACT + a];
      const float av = zz - q;
      out[OFF_AV + (size_t)m * ACT + a]  = av;  // z - sg(q)
      out[OFF_QST + (size_t)m * ACT + a] = q;   // z + sg(q - z) == q
      lsum += av * av;
    }
  }
#pragma unroll
  for (int msk = 1; msk < 32; msk <<= 1) lsum += __shfl_xor(lsum, msk, 32);
  if (lane == 0) loss_part[wave] = lsum;
  __syncthreads();
  if (tid == 0) {
    float t = 0.f;
    for (int w = 0; w < 8; ++w) t += loss_part[w];
    const float scale = 1.0f / ((float)MROW * (float)ACT);
    atomicAdd(&out[OFF_QL], t * scale);  // quant_loss
    atomicAdd(&out[OFF_CL], t * scale);  // commit_loss (same forward value)
  }
}

extern "C" void kernel_launch(void* const* d_in, const int* in_sizes, int n_in,
                              void* d_out, int out_size, void* d_ws, size_t ws_size,
                              hipStream_t stream) {
  const float* slots    = (const float*)d_in[0];
  const float* noise    = (const float*)d_in[1];
  const float* W1       = (const float*)d_in[2];
  const float* b1       = (const float*)d_in[3];
  const float* ln_g     = (const float*)d_in[4];
  const float* ln_b     = (const float*)d_in[5];
  const float* W2       = (const float*)d_in[6];
  const float* b2       = (const float*)d_in[7];
  const float* Wm       = (const float*)d_in[8];
  const float* bm       = (const float*)d_in[9];
  const float* Wv       = (const float*)d_in[10];
  const float* bv       = (const float*)d_in[11];
  const float* codebook = (const float*)d_in[12];
  float* out = (float*)d_out;

  float* mean_tok = (float*)d_ws;                       // T*ACT floats
  float* var_tok  = mean_tok + (size_t)TTOK * ACT;      // T*ACT floats

  init_loss_kernel<<<1, 32, 0, stream>>>(out);
  enc_kernel<<<TTOK / 16, 256, 0, stream>>>(slots, W1, b1, ln_g, ln_b, W2, b2,
                                            Wm, bm, Wv, bv, mean_tok, var_tok);
  vq_kernel<<<MROW / 16, 256, 0, stream>>>(noise, codebook, mean_tok, var_tok,
                                           out);
}